// DiscreteContinuousConvS2_79852031967248
// MI455X (gfx1250) — compile-verified
//
#include <hip/hip_runtime.h>

typedef __attribute__((ext_vector_type(2))) float v2f;
typedef __attribute__((ext_vector_type(8))) float v8f;

#define NLAT_IN  360
#define NLON_IN  720
#define NLAT_OUT 180
#define NLON_OUT 360
#define KSIZE    3
#define NSEG     (KSIZE * NLAT_OUT)   // 540 fused (kernel, nlat_out) segments
#define PTILE    16
#define CHUNK    512

// Kernel 1: CSR row pointers from sorted COO segment ids (lower_bound).
__global__ void build_row_ptr(const int* __restrict__ seg, int nnz,
                              int* __restrict__ row_ptr) {
    int s = blockIdx.x * blockDim.x + threadIdx.x;
    if (s > NSEG) return;
    int lo = 0, hi = nnz;
    while (lo < hi) {
        int mid = (lo + hi) >> 1;
        if (seg[mid] < s) lo = mid + 1; else hi = mid;
    }
    row_ptr[s] = lo;
}

// Kernel 2: one block per (segment, 16-wide p tile). 256 threads = 8 waves;
// each wave owns 16 channels. Accumulate M[p, bc] tile in WMMA A-layout
// (lane = p row, regs = bc cols), then transpose via D = A x I using
// v_wmma_f32_16x16x4_f32 so stores are p-contiguous b128 per channel.
__global__ __launch_bounds__(256) void disco_conv(
    const float* __restrict__ x,      // [128, 360, 720]
    const float* __restrict__ qw,     // [360]
    const int*   __restrict__ plat,
    const int*   __restrict__ plon,
    const float* __restrict__ pval,
    const int*   __restrict__ row_ptr,
    float*       __restrict__ out) {  // [128, 540, 360]
    __shared__ int   s_lat[CHUNK];
    __shared__ int   s_lon[CHUNK];
    __shared__ float s_wv[CHUNK];

    const int s   = blockIdx.x;
    const int p0  = blockIdx.y * PTILE;
    const int tid = threadIdx.x;
    const int wid = tid >> 5;        // wave id: channel group
    const int ln  = tid & 31;
    const int q   = ln & 15;         // p-within-tile (A-layout row) this lane owns
    const int hh  = ln >> 4;         // lane half selects K-pair / bc subset
    const int bc0 = wid * 16;

    // Per-lane channel base pointers: bc_local = 4c + 2*hh + j  (A-layout cols)
    const float* xb[8];
#pragma unroll
    for (int c = 0; c < 4; ++c)
#pragma unroll
        for (int j = 0; j < 2; ++j)
            xb[c * 2 + j] =
                x + (size_t)(bc0 + 4 * c + 2 * hh + j) * (NLAT_IN * NLON_IN);

    float a[8] = {0.f, 0.f, 0.f, 0.f, 0.f, 0.f, 0.f, 0.f};
    const int shift = 2 * (p0 + q + 1);   // in [2, 736]

    const int jb = row_ptr[s];
    const int je = row_ptr[s + 1];
    for (int j0 = jb; j0 < je; j0 += CHUNK) {
        const int cnt = (je - j0 < CHUNK) ? (je - j0) : CHUNK;
        __syncthreads();
        for (int t = tid; t < cnt; t += 256) {
            const int la = plat[j0 + t];
            s_lat[t] = la;
            s_lon[t] = plon[j0 + t];
            s_wv[t]  = pval[j0 + t] * qw[la];   // fold quadrature weight
        }
        __syncthreads();
        for (int u = 0; u < cnt; ++u) {
            const int   la = s_lat[u];
            const float w  = s_wv[u];
            int col = s_lon[u] - shift;         // >= -736
            col += (col < 0) ? NLON_IN : 0;
            col += (col < 0) ? NLON_IN : 0;     // now in [0, 720)
            const int off = la * NLON_IN + col; // L2-resident gathers
#pragma unroll
            for (int i = 0; i < 8; ++i)
                a[i] = fmaf(w, xb[i][off], a[i]);
        }
    }

    // In-register transpose: D[m,n] = sum_c A_c(16x4) * I_c(4x16) = M^T layout.
    // EXEC is all ones here (no divergence before this point at wave scope).
    v8f d = {0.f, 0.f, 0.f, 0.f, 0.f, 0.f, 0.f, 0.f};
    const int n  = ln & 15;   // output column (bc_local) this lane ends up with
    const int kb = 2 * hh;    // K-pair this lane half supplies
#pragma unroll
    for (int c = 0; c < 4; ++c) {
        v2f A; A[0] = a[2 * c]; A[1] = a[2 * c + 1];
        v2f Bm;
        Bm[0] = (4 * c + kb     == n) ? 1.0f : 0.0f;  // identity slice, B-layout
        Bm[1] = (4 * c + kb + 1 == n) ? 1.0f : 0.0f;
        d = __builtin_amdgcn_wmma_f32_16x16x4_f32(
                /*neg_a=*/false, A, /*neg_b=*/false, Bm,
                /*c_mod=*/(short)0, d, /*reuse_a=*/false, /*reuse_b=*/false);
    }

    // Lane now holds 8 consecutive p values (rows m = 8*hh + v) for one channel.
    const int bc = bc0 + n;
    const int qb = 8 * hh;
    if (p0 + qb < NLON_OUT) {   // mask upper half on the final partial p tile
        const size_t ob = (size_t)bc * ((size_t)NSEG * NLON_OUT)
                        + (size_t)s * NLON_OUT + (size_t)(p0 + qb);
        float4 f0 = make_float4(d[0], d[1], d[2], d[3]);
        float4 f1 = make_float4(d[4], d[5], d[6], d[7]);
        *(float4*)(out + ob)     = f0;   // 16B vector, 64B-contiguous per bc row
        *(float4*)(out + ob + 4) = f1;
    }
}

extern "C" void kernel_launch(void* const* d_in, const int* in_sizes, int n_in,
                              void* d_out, int out_size, void* d_ws, size_t ws_size,
                              hipStream_t stream) {
    const float* x   = (const float*)d_in[0];
    const float* qw  = (const float*)d_in[1];
    const int*   seg = (const int*)  d_in[2];
    const int*   lat = (const int*)  d_in[3];
    const int*   lon = (const int*)  d_in[4];
    const float* val = (const float*)d_in[5];
    float* out = (float*)d_out;
    const int nnz = in_sizes[2];

    int* row_ptr = (int*)d_ws;   // (NSEG+1) ints = 2164 B of scratch

    build_row_ptr<<<(NSEG + 1 + 255) / 256, 256, 0, stream>>>(seg, nnz, row_ptr);

    dim3 grid(NSEG, (NLON_OUT + PTILE - 1) / PTILE);   // 540 x 23 blocks
    disco_conv<<<grid, 256, 0, stream>>>(x, qw, lat, lon, val, row_ptr, out);
}